// IgniteGenerator_35433480192876
// MI455X (gfx1250) — compile-verified
//
#include <hip/hip_runtime.h>

typedef __attribute__((ext_vector_type(2))) float v2f;
typedef __attribute__((ext_vector_type(8))) float v8f;

#define ATT_SLOPE 0.2f
#define ACT_SLOPE 0.01f

// order-preserving float <-> uint mapping for atomic max on floats
__device__ __forceinline__ unsigned fkey(float f) {
  unsigned b = __float_as_uint(f);
  return (b & 0x80000000u) ? ~b : (b | 0x80000000u);
}
__device__ __forceinline__ float fdecode(unsigned k) {
  unsigned b = (k & 0x80000000u) ? (k & 0x7FFFFFFFu) : ~k;
  return __uint_as_float(b);
}

__global__ void k_zero(unsigned* __restrict__ p, int n) {
  int i = blockIdx.x * blockDim.x + threadIdx.x;
  if (i < n) p[i] = 0u;
}

// H = X[rows x 128] @ W[128 x 128], fp32 WMMA (v_wmma_f32_16x16x4_f32).
// Block = 256 threads = 8 waves; block handles a 16-row tile, wave w the cols [16w,16w+16).
__global__ void k_gemm128(const float* __restrict__ X, const float* __restrict__ W,
                          float* __restrict__ H, int rows) {
  const int lane = threadIdx.x & 31;
  const int wave = threadIdx.x >> 5;
  const int half = lane >> 4;   // 0 or 1
  const int l15  = lane & 15;
  int m = blockIdx.x * 16 + l15;
  if (m >= rows) m = rows - 1;  // clamp; rows % 16 == 0 in practice
  const int kb = half * 2;      // A/B K-pair held by this lane half
  const int n  = wave * 16 + l15;
  const float* __restrict__ xrow = X + (size_t)m * 128;
  v8f acc = {};
#pragma unroll
  for (int k0 = 0; k0 < 128; k0 += 4) {
    // A 16x4: lane<16 -> K=0,1 ; lane>=16 -> K=2,3 (M = lane&15)
    v2f a = *(const v2f*)(xrow + k0 + kb);
    // B 4x16: lane<16 -> K=0,1 ; lane>=16 -> K=2,3 (N = lane&15)
    v2f b;
    b.x = W[(k0 + kb) * 128 + n];
    b.y = W[(k0 + kb + 1) * 128 + n];
    acc = __builtin_amdgcn_wmma_f32_16x16x4_f32(false, a, false, b, (short)0, acc,
                                                false, false);
  }
  // C/D 16x16: VGPR r holds M = r + 8*(lane>=16), N = lane&15
  const int rbase = blockIdx.x * 16 + half * 8;
#pragma unroll
  for (int r = 0; r < 8; ++r) {
    int row = rbase + r;
    if (row < rows) H[(size_t)row * 128 + n] = acc[r];
  }
}

// al_src[n,h] = sum_d H[n, h*16+d]*a_src[h,d]; same for al_dst. One thread per (node,head).
__global__ void k_attn_logits(const float* __restrict__ H,
                              const float* __restrict__ a_src, const float* __restrict__ a_dst,
                              float* __restrict__ alS, float* __restrict__ alD, int n_nodes) {
  int i = blockIdx.x * blockDim.x + threadIdx.x;
  if (i >= n_nodes * 8) return;
  int hh = i & 7;
  int node = i >> 3;
  const float* __restrict__ hp = H + (size_t)node * 128 + hh * 16;
  const float* __restrict__ as = a_src + hh * 16;
  const float* __restrict__ ad = a_dst + hh * 16;
  float s = 0.f, d = 0.f;
#pragma unroll
  for (int j = 0; j < 16; ++j) { float hv = hp[j]; s += hv * as[j]; d += hv * ad[j]; }
  alS[i] = s;
  alD[i] = d;
}

// Pass A: per-(edge,head) attention logit -> atomic max over dst segments.
__global__ void k_edge_max(const int* __restrict__ src, const int* __restrict__ dst,
                           int n_edges, int n_nodes,
                           const float* __restrict__ alS, const float* __restrict__ alD,
                           unsigned* __restrict__ mkey) {
  int i = blockIdx.x * blockDim.x + threadIdx.x;
  int total = (n_edges + n_nodes) * 8;
  if (i >= total) return;
  int h = i & 7;
  int e = i >> 3;
  int s, d;
  if (e < n_edges) { s = src[e]; d = dst[e]; } else { s = d = e - n_edges; }
  float v = alS[s * 8 + h] + alD[d * 8 + h];
  v = v > 0.f ? v : ATT_SLOPE * v;
  atomicMax(&mkey[d * 8 + h], fkey(v));
}

// Pass B: one wave32 per edge. Lanes 0..7 compute ew per head (+atomic s),
// all 32 lanes scatter ew * h[src] (4 channels each) into agg[dst] via atomics.
__global__ void k_edge_agg(const int* __restrict__ src, const int* __restrict__ dst,
                           int n_edges, int n_nodes,
                           const float* __restrict__ alS, const float* __restrict__ alD,
                           const unsigned* __restrict__ mkey, float* __restrict__ ssum,
                           const float* __restrict__ H, float* __restrict__ agg) {
  int lane = threadIdx.x & 31;
  int wave = threadIdx.x >> 5;
  int e = blockIdx.x * 8 + wave;
  int total = n_edges + n_nodes;
  if (e >= total) return;   // uniform per wave
  int s, d;
  if (e < n_edges) { s = src[e]; d = dst[e]; } else { s = d = e - n_edges; }
  float ew8 = 0.f;
  if (lane < 8) {
    float v = alS[s * 8 + lane] + alD[d * 8 + lane];
    v = v > 0.f ? v : ATT_SLOPE * v;
    float m = fdecode(mkey[d * 8 + lane]);
    ew8 = expf(v - m);
    atomicAdd(&ssum[d * 8 + lane], ew8);
  }
  float ew = __shfl(ew8, lane >> 2, 32);  // head of channels [4*lane,4*lane+4) is lane>>2
  const float4 hv = *(const float4*)(H + (size_t)s * 128 + lane * 4);
  float* ap = agg + (size_t)d * 128 + lane * 4;
  atomicAdd(ap + 0, hv.x * ew);
  atomicAdd(ap + 1, hv.y * ew);
  atomicAdd(ap + 2, hv.z * ew);
  atomicAdd(ap + 3, hv.w * ew);
}

// Pass C: out[n,c] = (relu?)( agg[n,c] / (s[n,head]+1e-16) + bias[c] ), in place.
__global__ void k_norm(float* __restrict__ agg, const float* __restrict__ ssum,
                       const float* __restrict__ bias, int n_nodes, int do_relu) {
  int i = blockIdx.x * blockDim.x + threadIdx.x;
  if (i >= n_nodes * 128) return;
  int c = i & 127;
  int node = i >> 7;
  float sv = ssum[node * 8 + (c >> 4)] + 1e-16f;
  float v = agg[i] / sv + bias[c];
  if (do_relu) v = v > 0.f ? v : 0.f;
  agg[i] = v;
}

// Final heads: wave per gathered row; two 128-dots via lane partials + shfl reduce.
__global__ void k_heads(const float* __restrict__ Z, const int* __restrict__ tidx,
                        const int* __restrict__ cidx, int nT, int nC,
                        const float* __restrict__ wy1, const float* __restrict__ by1,
                        const float* __restrict__ wy0, const float* __restrict__ by0,
                        float* __restrict__ out) {
  int lane = threadIdx.x & 31;
  int wave = threadIdx.x >> 5;
  int row = blockIdx.x * 8 + wave;
  if (row >= nT + nC) return;
  int node = (row < nT) ? tidx[row] : cidx[row - nT];
  const float4 z  = *(const float4*)(Z + (size_t)node * 128 + lane * 4);
  const float4 w1 = *(const float4*)(wy1 + lane * 4);
  const float4 w0 = *(const float4*)(wy0 + lane * 4);
  float d1 = z.x * w1.x + z.y * w1.y + z.z * w1.z + z.w * w1.w;
  float d0 = z.x * w0.x + z.y * w0.y + z.z * w0.z + z.w * w0.w;
#pragma unroll
  for (int off = 16; off > 0; off >>= 1) {
    d1 += __shfl_xor(d1, off, 32);
    d0 += __shfl_xor(d0, off, 32);
  }
  if (lane == 0) {
    float v1 = d1 + by1[0]; v1 = v1 > 0.f ? v1 : ACT_SLOPE * v1;
    float v0 = d0 + by0[0]; v0 = v0 > 0.f ? v0 : ACT_SLOPE * v0;
    if (row < nT) { out[row] = v1; out[nT + row] = v0; }
    else { int i2 = row - nT; out[2 * nT + i2] = v0; out[2 * nT + nC + i2] = v1; }
  }
}

extern "C" void kernel_launch(void* const* d_in, const int* in_sizes, int n_in,
                              void* d_out, int out_size, void* d_ws, size_t ws_size,
                              hipStream_t stream) {
  const float* x   = (const float*)d_in[0];
  const int*   ei  = (const int*)d_in[1];
  const int*   tix = (const int*)d_in[2];
  const int*   cix = (const int*)d_in[3];
  const float* W1  = (const float*)d_in[4];
  const float* as1 = (const float*)d_in[5];
  const float* ad1 = (const float*)d_in[6];
  const float* b1  = (const float*)d_in[7];
  const float* W2  = (const float*)d_in[8];
  const float* as2 = (const float*)d_in[9];
  const float* ad2 = (const float*)d_in[10];
  const float* b2  = (const float*)d_in[11];
  const float* wy1 = (const float*)d_in[12];
  const float* by1 = (const float*)d_in[13];
  const float* wy0 = (const float*)d_in[14];
  const float* by0 = (const float*)d_in[15];

  const int N  = in_sizes[0] / 128;
  const int E  = in_sizes[1] / 2;
  const int nT = in_sizes[2];
  const int nC = in_sizes[3];
  const int* src = ei;
  const int* dst = ei + E;

  float* ws   = (float*)d_ws;
  float* h    = ws;                          // N*128 (pre-agg features)
  float* xZ1  = ws + (size_t)N * 128;        // N*128 (layer-1 agg, then xZ1)
  float* alS  = xZ1 + (size_t)N * 128;       // N*8
  float* alD  = alS + (size_t)N * 8;         // N*8
  unsigned* mkey = (unsigned*)(alD + (size_t)N * 8);  // N*8
  float* ssum = (float*)(mkey + (size_t)N * 8);       // N*8

  float* out = (float*)d_out;
  float* xZ2 = out + 2 * (size_t)(nT + nC);  // layer-2 agg -> final xZ2 (in d_out)

  const int total_e = E + N;
  dim3 blk(256);
  auto cdiv = [](long a, long b) { return (int)((a + b - 1) / b); };

  // ---------------- layer 1 ----------------
  k_gemm128<<<cdiv(N, 16), blk, 0, stream>>>(x, W1, h, N);
  k_attn_logits<<<cdiv((long)N * 8, 256), blk, 0, stream>>>(h, as1, ad1, alS, alD, N);
  k_zero<<<cdiv((long)N * 128, 256), blk, 0, stream>>>((unsigned*)xZ1, N * 128);
  k_zero<<<cdiv((long)N * 8, 256), blk, 0, stream>>>(mkey, N * 8);
  k_zero<<<cdiv((long)N * 8, 256), blk, 0, stream>>>((unsigned*)ssum, N * 8);
  k_edge_max<<<cdiv((long)total_e * 8, 256), blk, 0, stream>>>(src, dst, E, N, alS, alD, mkey);
  k_edge_agg<<<cdiv(total_e, 8), blk, 0, stream>>>(src, dst, E, N, alS, alD, mkey, ssum, h, xZ1);
  k_norm<<<cdiv((long)N * 128, 256), blk, 0, stream>>>(xZ1, ssum, b1, N, 1);

  // ---------------- layer 2 ----------------
  k_gemm128<<<cdiv(N, 16), blk, 0, stream>>>(xZ1, W2, h, N);
  k_attn_logits<<<cdiv((long)N * 8, 256), blk, 0, stream>>>(h, as2, ad2, alS, alD, N);
  k_zero<<<cdiv((long)N * 128, 256), blk, 0, stream>>>((unsigned*)xZ2, N * 128);
  k_zero<<<cdiv((long)N * 8, 256), blk, 0, stream>>>(mkey, N * 8);
  k_zero<<<cdiv((long)N * 8, 256), blk, 0, stream>>>((unsigned*)ssum, N * 8);
  k_edge_max<<<cdiv((long)total_e * 8, 256), blk, 0, stream>>>(src, dst, E, N, alS, alD, mkey);
  k_edge_agg<<<cdiv(total_e, 8), blk, 0, stream>>>(src, dst, E, N, alS, alD, mkey, ssum, h, xZ2);
  k_norm<<<cdiv((long)N * 128, 256), blk, 0, stream>>>(xZ2, ssum, b2, N, 0);

  // ---------------- outcome heads ----------------
  k_heads<<<cdiv(nT + nC, 8), blk, 0, stream>>>(xZ2, tix, cix, nT, nC, wy1, by1, wy0, by0, out);
}